// GCN_68959994904704
// MI455X (gfx1250) — compile-verified
//
#include <hip/hip_runtime.h>
#include <math.h>

// CDNA5 / gfx1250: wave32, WMMA (not MFMA). Full-precision f32 WMMA 16x16x4.
typedef __attribute__((ext_vector_type(2))) float v2f;
typedef __attribute__((ext_vector_type(8))) float v8f;

#define NC    45    // input feature dim
#define NCP   48    // padded K for layer-1 WMMA (12 chunks of 4)
#define H1DIM 32    // layer-1 / hidden dim
#define H2DIM 128   // layer-2 output dim

// -------------------------------------------------------------------------
// deg[i] = sum of ew over edges with dst==i  (self-loop +1 added later)
__global__ void deg_kernel(const int* __restrict__ dst, const float* __restrict__ ew,
                           float* __restrict__ deg, int E) {
  int e = blockIdx.x * blockDim.x + threadIdx.x;
  if (e < E) atomicAdd(&deg[dst[e]], ew[e]);
}

// dis[i] = 1/sqrt(deg[i] + 1); deg+1 >= 1 so no zero-branch needed
__global__ void dis_kernel(float* __restrict__ deg, int n) {
  int i = blockIdx.x * blockDim.x + threadIdx.x;
  if (i < n) deg[i] = 1.0f / sqrtf(deg[i] + 1.0f);
}

// -------------------------------------------------------------------------
// Operand prep: zero-pad x to K=48 and pack W1/W2 into per-lane fragment
// order so the WMMA loops contain only unconditional b64 loads (no exec-mask
// branches around loads, which bloated the previous build).
__global__ void pad_x_kernel(const float* __restrict__ x, float* __restrict__ xp,
                             int total /* n*48 */) {
  int gid = blockIdx.x * blockDim.x + threadIdx.x;
  if (gid >= total) return;
  int i = gid / NCP, c = gid - i * NCP;
  xp[gid] = (c < NC) ? x[(size_t)i * NC + c] : 0.0f;
}

// Bp1[((k*2+kh)*32 + col)*2 + p] = W1[(4k+2kh+p)*32 + col], zero-padded rows>=45
__global__ void pack_w1_kernel(const float* __restrict__ W1, float* __restrict__ Bp1) {
  int gid = blockIdx.x * blockDim.x + threadIdx.x;
  if (gid >= 12 * 2 * H1DIM * 2) return;
  int p   = gid & 1;
  int col = (gid >> 1) & 31;
  int kh  = (gid >> 6) & 1;
  int k   = gid >> 7;
  int row = 4 * k + 2 * kh + p;
  Bp1[gid] = (row < NC) ? W1[row * H1DIM + col] : 0.0f;
}

// Bp2[((k*2+kh)*128 + col)*2 + p] = W2[(4k+2kh+p)*128 + col]  (K=32 exact)
__global__ void pack_w2_kernel(const float* __restrict__ W2, float* __restrict__ Bp2) {
  int gid = blockIdx.x * blockDim.x + threadIdx.x;
  if (gid >= 8 * 2 * H2DIM * 2) return;
  int p   = gid & 1;
  int col = (gid >> 1) & 127;
  int kh  = (gid >> 8) & 1;
  int k   = gid >> 9;
  int row = 4 * k + 2 * kh + p;
  Bp2[gid] = W2[row * H2DIM + col];
}

// -------------------------------------------------------------------------
// GEMM1: h1[n x 32] = xp[n x 48] @ W1p. One wave per 16-row stripe.
// ISA 7.12.2 fp32 layouts: A 16x4 lane%16=M, lanes>=16 carry K={2,3} (VGPR=K%2);
// B 4x16 lane%16=N, same K striping; C VGPR j -> M=j+8*(lane>=16), N=lane%16.
__global__ void gemm1_wmma(const float* __restrict__ xp, const float* __restrict__ Bp1,
                           float* __restrict__ h1, int n) {
  const int lane  = threadIdx.x;       // block == 1 wave, EXEC all ones
  const int m0    = blockIdx.x * 16;
  const int khalf = lane >> 4;
  const int nn    = lane & 15;
  int m = m0 + nn;
  if (m > n - 1) m = n - 1;

  const float* ap = xp + (size_t)m * NCP + 2 * khalf;     // +4k per chunk
  const float* bp = Bp1 + khalf * 64;                     // +128 per chunk
  v8f c0 = {};
  v8f c1 = {};
#pragma unroll
  for (int k = 0; k < 12; ++k) {
    v2f a  = *(const v2f*)(ap + 4 * k);
    v2f b0 = *(const v2f*)(bp + 128 * k + nn * 2);
    v2f b1 = *(const v2f*)(bp + 128 * k + (16 + nn) * 2);
    c0 = __builtin_amdgcn_wmma_f32_16x16x4_f32(false, a, false, b0, (short)0, c0, false, false);
    c1 = __builtin_amdgcn_wmma_f32_16x16x4_f32(false, a, false, b1, (short)0, c1, false, false);
  }
  const int rbase = m0 + (khalf << 3);
#pragma unroll
  for (int j = 0; j < 8; ++j) {
    const int r = rbase + j;
    if (r < n) {
      h1[(size_t)r * H1DIM + nn]      = c0[j];
      h1[(size_t)r * H1DIM + 16 + nn] = c1[j];
    }
  }
}

// -------------------------------------------------------------------------
// GEMM2: out[n x 128] = g[n x 32] @ W2 + b2, streamed straight to d_out
// (aggregation was hoisted BEFORE this GEMM: A(gW2) == (Ag)W2, so the edge
// scatter runs on 32 features instead of 128 -> 4x less atomic traffic).
__global__ void gemm2_wmma(const float* __restrict__ g, const float* __restrict__ Bp2,
                           const float* __restrict__ b2, float* __restrict__ out, int n) {
  const int lane  = threadIdx.x;
  const int m0    = blockIdx.x * 16;
  const int khalf = lane >> 4;
  const int nn    = lane & 15;
  int m = m0 + nn;
  if (m > n - 1) m = n - 1;

  const float* ap = g + (size_t)m * H1DIM + 2 * khalf;    // +4k per chunk
  const float* bp = Bp2 + khalf * 256;                    // +512 per chunk
  v2f a[8];
#pragma unroll
  for (int k = 0; k < 8; ++k) a[k] = *(const v2f*)(ap + 4 * k);

  v8f acc[8] = {};
#pragma unroll
  for (int k = 0; k < 8; ++k) {
#pragma unroll
    for (int t = 0; t < 8; ++t) {
      v2f b = *(const v2f*)(bp + 512 * k + (t * 16 + nn) * 2);
      acc[t] = __builtin_amdgcn_wmma_f32_16x16x4_f32(false, a[k], false, b, (short)0, acc[t], false, false);
    }
  }
  float bv[8];
#pragma unroll
  for (int t = 0; t < 8; ++t) bv[t] = b2[t * 16 + nn];

  const int rbase = m0 + (khalf << 3);
#pragma unroll
  for (int j = 0; j < 8; ++j) {
    const int r = rbase + j;
    if (r < n) {
#pragma unroll
      for (int t = 0; t < 8; ++t)
        out[(size_t)r * H2DIM + t * 16 + nn] = acc[t][j] + bv[t];
    }
  }
}

// -------------------------------------------------------------------------
// Edge scatter-add over 32 features: 8 lanes per edge, float4 gather + 4 f32
// atomics each. h / agg are L2-resident (12.8MB each vs 192MB L2).
__global__ void agg_edges_kernel(const int* __restrict__ src, const int* __restrict__ dst,
                                 const float* __restrict__ ew, const float* __restrict__ dis,
                                 const float* __restrict__ h, float* __restrict__ agg, int E) {
  long long gid = (long long)blockIdx.x * blockDim.x + threadIdx.x;
  int e = (int)(gid >> 3);
  int q = (int)(gid & 7) << 2;
  if (e >= E) return;
  int s = src[e], d = dst[e];
  float norm = dis[s] * ew[e] * dis[d];
  const float4 v = *(const float4*)(h + (size_t)s * H1DIM + q);
  float* o = agg + (size_t)d * H1DIM + q;
  atomicAdd(o + 0, v.x * norm);
  atomicAdd(o + 1, v.y * norm);
  atomicAdd(o + 2, v.z * norm);
  atomicAdd(o + 3, v.w * norm);
}

// h = tanh(agg1 + h1*dis^2 + b1); self-loop (norm = dis^2) hoisted out of the
// edge loop. In-place into agg1.
__global__ void tanh_bias_kernel(float* __restrict__ agg, const float* __restrict__ h1,
                                 const float* __restrict__ dis, const float* __restrict__ b1,
                                 int n) {
  int gid = blockIdx.x * blockDim.x + threadIdx.x;
  if (gid >= n * H1DIM) return;
  int i = gid >> 5, c = gid & 31;
  float di = dis[i];
  agg[gid] = tanhf(agg[gid] + h1[gid] * di * di + b1[c]);
}

// g = agg2 + h*dis^2  (layer-2 self-loop), in-place into agg2
__global__ void selfloop_kernel(float* __restrict__ agg2, const float* __restrict__ h,
                                const float* __restrict__ dis, int n) {
  int gid = blockIdx.x * blockDim.x + threadIdx.x;
  if (gid >= n * H1DIM) return;
  int i = gid >> 5;
  float di = dis[i];
  agg2[gid] += h[gid] * di * di;
}

// -------------------------------------------------------------------------
extern "C" void kernel_launch(void* const* d_in, const int* in_sizes, int n_in,
                              void* d_out, int out_size, void* d_ws, size_t ws_size,
                              hipStream_t stream) {
  const float* x   = (const float*)d_in[0];
  const int*   src = (const int*)  d_in[1];
  const int*   dst = (const int*)  d_in[2];
  const float* ew  = (const float*)d_in[3];
  const float* W1  = (const float*)d_in[4];
  const float* b1  = (const float*)d_in[5];
  const float* W2  = (const float*)d_in[6];
  const float* b2  = (const float*)d_in[7];
  float* out = (float*)d_out;

  const int n = in_sizes[0] / NC;   // 100000 (divisible by 16)
  const int E = in_sizes[1];        // 1600000

  // Workspace (floats): dis n | xp 48n | h1 32n | agg1 32n | agg2 32n | Bp1 | Bp2
  // ~58MB. All sub-buffer offsets are multiples of n*4B = 400000B (16B aligned).
  float* dis  = (float*)d_ws;
  float* xp   = dis  + n;
  float* h1   = xp   + (size_t)n * NCP;
  float* agg1 = h1   + (size_t)n * H1DIM;
  float* agg2 = agg1 + (size_t)n * H1DIM;
  float* Bp1  = agg2 + (size_t)n * H1DIM;
  float* Bp2  = Bp1  + 12 * 2 * H1DIM * 2;

  // Zero accumulators (capture-safe async memsets). d_out needs no zeroing:
  // gemm2 writes every element exactly once.
  hipMemsetAsync(dis,  0, (size_t)n * sizeof(float), stream);
  hipMemsetAsync(agg1, 0, (size_t)n * H1DIM * sizeof(float), stream);
  hipMemsetAsync(agg2, 0, (size_t)n * H1DIM * sizeof(float), stream);

  const int T = 256;
  // Shared symmetric normalization (identical for both layers)
  deg_kernel<<<(E + T - 1) / T, T, 0, stream>>>(dst, ew, dis, E);
  dis_kernel<<<(n + T - 1) / T, T, 0, stream>>>(dis, n);

  // Operand prep for branch-free WMMA loops
  pad_x_kernel<<<((long long)n * NCP + T - 1) / T, T, 0, stream>>>(x, xp, n * NCP);
  pack_w1_kernel<<<(12 * 2 * H1DIM * 2 + T - 1) / T, T, 0, stream>>>(W1, Bp1);
  pack_w2_kernel<<<(8 * 2 * H2DIM * 2 + T - 1) / T, T, 0, stream>>>(W2, Bp2);

  // Layer 1: transform (WMMA) -> edge aggregate (32 feat) -> tanh/bias/selfloop
  gemm1_wmma<<<(n + 15) / 16, 32, 0, stream>>>(xp, Bp1, h1, n);
  {
    long long work = (long long)E * 8;
    agg_edges_kernel<<<(unsigned)((work + T - 1) / T), T, 0, stream>>>(src, dst, ew, dis, h1, agg1, E);
  }
  tanh_bias_kernel<<<((long long)n * H1DIM + T - 1) / T, T, 0, stream>>>(agg1, h1, dis, b1, n);

  // Layer 2: aggregate FIRST on 32 features (A(hW2) == (Ah)W2), then WMMA
  // transform streams the 128-wide result + bias directly into d_out.
  {
    long long work = (long long)E * 8;
    agg_edges_kernel<<<(unsigned)((work + T - 1) / T), T, 0, stream>>>(src, dst, ew, dis, agg1, agg2, E);
  }
  selfloop_kernel<<<((long long)n * H1DIM + T - 1) / T, T, 0, stream>>>(agg2, agg1, dis, n);
  gemm2_wmma<<<(n + 15) / 16, 32, 0, stream>>>(agg2, Bp2, b2, out, n);
}